// SimpleGIN_61040075211351
// MI455X (gfx1250) — compile-verified
//
#include <hip/hip_runtime.h>
#include <hip/hip_bf16.h>
#include <math.h>

// ---------------------------------------------------------------------------
// SimpleGIN forward for MI455X (gfx1250, wave32, WMMA).
//
// Pipeline (all fp32, matching the fp32 reference — this net is aggregation-
// bound, not FLOP-bound, so fp32 WMMA (V_WMMA_F32_16X16X4_F32) costs nothing
// vs f16 and keeps numerics exact):
//   agg  = segment_sum(x[src], dst)                       (atomic scatter)
//   h1   = relu( relu(((1+eps1)x + agg) W11 + b11) W12 + b12 )
//   agg  = segment_sum(h1[src], dst)
//   h2   = relu( relu(((1+eps2)h1 + agg) W21 + b21) W22 + b22 )
//   out  = log_softmax(h2 Wl + bl)
// ---------------------------------------------------------------------------

#define NN   50000
#define EE   800000
#define DIM  128
#define OUTD 64
#define ZPAD 132   // 128 + 4: row pad -> conflict-free column reads from LDS
#define LPAD 68    // 64 + 4

typedef __attribute__((ext_vector_type(2))) float v2f;
typedef __attribute__((ext_vector_type(8))) float v8f;

// ---------------------------------------------------------------------------
__global__ void __launch_bounds__(256)
gin_zero_kernel(float4* __restrict__ p, int n4) {
    int i = blockIdx.x * 256 + threadIdx.x;
    if (i < n4) p[i] = make_float4(0.f, 0.f, 0.f, 0.f);
}

// 32 threads per edge, each handles 4 contiguous features.
__global__ void __launch_bounds__(256)
gin_scatter_kernel(const float* __restrict__ h,
                   const int* __restrict__ src,
                   const int* __restrict__ dst,
                   float* __restrict__ agg) {
    long long t = (long long)blockIdx.x * 256 + threadIdx.x;
    int e = (int)(t >> 5);
    if (e >= EE) return;
    int f = ((int)t & 31) * 4;
    int s = src[e], d = dst[e];
    float4 v = *(const float4*)&h[(long long)s * DIM + f];
    float* a = &agg[(long long)d * DIM + f];
    unsafeAtomicAdd(a + 0, v.x);   // -> global_atomic_add_f32
    unsafeAtomicAdd(a + 1, v.y);
    unsafeAtomicAdd(a + 2, v.z);
    unsafeAtomicAdd(a + 3, v.w);
}

// ---------------------------------------------------------------------------
// Fused GIN MLP: one block = 16 node rows. 8 waves, each owning a 16-col tile.
// out = relu( relu(Z @ W1 + b1) @ W2 + b2 ),  Z = (1+eps)*h + agg
// ---------------------------------------------------------------------------
__global__ void __launch_bounds__(256)
gin_mlp_kernel(const float* __restrict__ h,  const float* __restrict__ agg,
               const float* __restrict__ eps_p,
               const float* __restrict__ W1, const float* __restrict__ b1,
               const float* __restrict__ W2, const float* __restrict__ b2,
               float* __restrict__ out) {
    __shared__ float Z[16][ZPAD];
    __shared__ float T[16][ZPAD];

    const int rowBase = blockIdx.x * 16;
    const int tid  = threadIdx.x;          // 0..255
    const float ep = 1.0f + eps_p[0];

    // Stage Z tile: 16x128 floats, 256 threads x 2 float4 each.
    for (int i = tid; i < 16 * (DIM / 4); i += 256) {
        int r  = i / (DIM / 4);
        int c4 = (i % (DIM / 4)) * 4;
        float4 hv = *(const float4*)&h  [(long long)(rowBase + r) * DIM + c4];
        float4 av = *(const float4*)&agg[(long long)(rowBase + r) * DIM + c4];
        float4 z;
        z.x = ep * hv.x + av.x;  z.y = ep * hv.y + av.y;
        z.z = ep * hv.z + av.z;  z.w = ep * hv.w + av.w;
        *(float4*)&Z[r][c4] = z;
    }
    __syncthreads();

    const int wave = tid >> 5;             // 0..7 -> column tile
    const int lane = tid & 31;
    const int n0   = wave * 16;
    const int lm   = lane & 15;
    const int koff = (lane < 16) ? 0 : 2;  // f32 16x16x4 A/B lane split
    const int mh   = (lane < 16) ? 0 : 8;  // C/D row offset per lane half

    // ---- GEMM1: T = relu(Z @ W1 + b1) ----
    v8f acc = {};
    #pragma unroll 4
    for (int k = 0; k < DIM; k += 4) {
        v2f a, b;
        a.x = Z[lm][k + koff];
        a.y = Z[lm][k + koff + 1];
        b.x = W1[(k + koff)     * DIM + n0 + lm];
        b.y = W1[(k + koff + 1) * DIM + n0 + lm];
        acc = __builtin_amdgcn_wmma_f32_16x16x4_f32(
                  false, a, false, b, (short)0, acc, false, false);
    }
    {
        float bias = b1[n0 + lm];
        #pragma unroll
        for (int r = 0; r < 8; ++r)
            T[r + mh][n0 + lm] = fmaxf(acc[r] + bias, 0.0f);
    }
    __syncthreads();

    // ---- GEMM2: out = relu(T @ W2 + b2) ----
    v8f acc2 = {};
    #pragma unroll 4
    for (int k = 0; k < DIM; k += 4) {
        v2f a, b;
        a.x = T[lm][k + koff];
        a.y = T[lm][k + koff + 1];
        b.x = W2[(k + koff)     * DIM + n0 + lm];
        b.y = W2[(k + koff + 1) * DIM + n0 + lm];
        acc2 = __builtin_amdgcn_wmma_f32_16x16x4_f32(
                   false, a, false, b, (short)0, acc2, false, false);
    }
    {
        float bias = b2[n0 + lm];
        #pragma unroll
        for (int r = 0; r < 8; ++r)
            out[(long long)(rowBase + r + mh) * DIM + n0 + lm] =
                fmaxf(acc2[r] + bias, 0.0f);
    }
}

// ---------------------------------------------------------------------------
// Head: logits = h2 @ Wl + bl  (16x64 tile per block), then log_softmax.
// ---------------------------------------------------------------------------
__global__ void __launch_bounds__(128)
gin_head_kernel(const float* __restrict__ h,
                const float* __restrict__ Wl, const float* __restrict__ bl,
                float* __restrict__ out) {
    __shared__ float Ht[16][ZPAD];
    __shared__ float L [16][LPAD];

    const int rowBase = blockIdx.x * 16;
    const int tid  = threadIdx.x;          // 0..127
    const int wave = tid >> 5;             // 0..3 -> column tile of 64
    const int lane = tid & 31;
    const int n0   = wave * 16;
    const int lm   = lane & 15;
    const int koff = (lane < 16) ? 0 : 2;
    const int mh   = (lane < 16) ? 0 : 8;

    for (int i = tid; i < 16 * (DIM / 4); i += 128) {
        int r  = i / (DIM / 4);
        int c4 = (i % (DIM / 4)) * 4;
        *(float4*)&Ht[r][c4] =
            *(const float4*)&h[(long long)(rowBase + r) * DIM + c4];
    }
    __syncthreads();

    v8f acc = {};
    #pragma unroll 4
    for (int k = 0; k < DIM; k += 4) {
        v2f a, b;
        a.x = Ht[lm][k + koff];
        a.y = Ht[lm][k + koff + 1];
        b.x = Wl[(k + koff)     * OUTD + n0 + lm];
        b.y = Wl[(k + koff + 1) * OUTD + n0 + lm];
        acc = __builtin_amdgcn_wmma_f32_16x16x4_f32(
                  false, a, false, b, (short)0, acc, false, false);
    }
    {
        float bias = bl[n0 + lm];
        #pragma unroll
        for (int r = 0; r < 8; ++r)
            L[r + mh][n0 + lm] = acc[r] + bias;
    }
    __syncthreads();

    // log_softmax over 64 cols; divergent section only after all WMMAs.
    if (tid < 16) {
        float mx = -INFINITY;
        for (int c = 0; c < OUTD; ++c) mx = fmaxf(mx, L[tid][c]);
        float s = 0.0f;
        for (int c = 0; c < OUTD; ++c) s += __expf(L[tid][c] - mx);
        float lse = mx + __logf(s);
        for (int c = 0; c < OUTD; ++c)
            out[(long long)(rowBase + tid) * OUTD + c] = L[tid][c] - lse;
    }
}

// ---------------------------------------------------------------------------
extern "C" void kernel_launch(void* const* d_in, const int* in_sizes, int n_in,
                              void* d_out, int out_size, void* d_ws, size_t ws_size,
                              hipStream_t stream) {
    (void)in_sizes; (void)n_in; (void)out_size; (void)ws_size;

    const float* x    = (const float*)d_in[0];
    const int*   ei   = (const int*)  d_in[1];   // [2,E] flat
    const float* eps1 = (const float*)d_in[2];
    const float* W11  = (const float*)d_in[3];
    const float* b11  = (const float*)d_in[4];
    const float* W12  = (const float*)d_in[5];
    const float* b12  = (const float*)d_in[6];
    const float* eps2 = (const float*)d_in[7];
    const float* W21  = (const float*)d_in[8];
    const float* b21  = (const float*)d_in[9];
    const float* W22  = (const float*)d_in[10];
    const float* b22  = (const float*)d_in[11];
    const float* Wl   = (const float*)d_in[12];
    const float* bl   = (const float*)d_in[13];
    float* out = (float*)d_out;

    const int* src = ei;
    const int* dst = ei + EE;

    // Workspace partition: agg | h1 | h2   (each N*128 fp32)
    float* agg = (float*)d_ws;
    float* h1  = agg + (size_t)NN * DIM;
    float* h2  = h1  + (size_t)NN * DIM;

    const int n4         = NN * DIM / 4;
    const int zeroBlocks = (n4 + 255) / 256;
    const int scatBlocks = (int)(((long long)EE * 32 + 255) / 256);
    const int tileBlocks = NN / 16;        // 3125, exact

    // ---- Layer 1 ----
    gin_zero_kernel<<<zeroBlocks, 256, 0, stream>>>((float4*)agg, n4);
    gin_scatter_kernel<<<scatBlocks, 256, 0, stream>>>(x, src, dst, agg);
    gin_mlp_kernel<<<tileBlocks, 256, 0, stream>>>(x, agg, eps1,
                                                   W11, b11, W12, b12, h1);
    // ---- Layer 2 ----
    gin_zero_kernel<<<zeroBlocks, 256, 0, stream>>>((float4*)agg, n4);
    gin_scatter_kernel<<<scatBlocks, 256, 0, stream>>>(h1, src, dst, agg);
    gin_mlp_kernel<<<tileBlocks, 256, 0, stream>>>(h1, agg, eps2,
                                                   W21, b21, W22, b22, h2);
    // ---- Head + log_softmax ----
    gin_head_kernel<<<tileBlocks, 128, 0, stream>>>(h2, Wl, bl, out);
}